// Decoder_15753940042227
// MI455X (gfx1250) — compile-verified
//
#include <hip/hip_runtime.h>
#include <stdint.h>

#define BB   16
#define TT   256
#define NTR  1024
#define NTE  256
#define KK1  16

// ---- CDNA5 async global->LDS copy (ASYNCcnt path) ----------------------
typedef __attribute__((ext_vector_type(4))) int v4i;
typedef __attribute__((address_space(1))) v4i* gbl_v4i_p;
typedef __attribute__((address_space(3))) v4i* lds_v4i_p;

__device__ __forceinline__ void async_copy_b128(const float* g, float* l) {
    __builtin_amdgcn_global_load_async_to_lds_b128(
        (gbl_v4i_p)(uintptr_t)g,
        (lds_v4i_p)(uint32_t)(uintptr_t)l,
        /*imm offset*/0, /*cpol*/0);
}

__device__ __forceinline__ void wait_async_all() {
#if __has_builtin(__builtin_amdgcn_s_wait_asynccnt)
    __builtin_amdgcn_s_wait_asynccnt(0);
#else
    asm volatile("s_wait_asynccnt 0" ::: "memory");
#endif
}

// One block per neuron n; 256 threads (8 waves, wave32). Computes out[b][n][t]
// for all b,t. z tensors staged once per block into LDS via async-DMA.
__global__ __launch_bounds__(256) void decoder_fused_kernel(
    const float* __restrict__ z0,       // (B,T,1)
    const float* __restrict__ z1,       // (B,T,2)
    const float* __restrict__ coeff0,   // (1,1)
    const float* __restrict__ mean0,    // (1,1,1)
    const float* __restrict__ log_var0, // (1,1,1)
    const float* __restrict__ coeff1,   // (1,16)
    const float* __restrict__ mean1,    // (1,16,2)
    const float* __restrict__ log_var1, // (1,16,2)
    const float* __restrict__ rf0,      // (N,1)
    const float* __restrict__ rf1,      // (N,2)
    const float* __restrict__ ew,       // (N,2)
    const float* __restrict__ lfs,      // (N,)
    float* __restrict__ out,            // (B,N,T)
    int N)
{
    __shared__ float s_z0[BB * TT];          // 16 KB
    __shared__ float s_z1[BB * TT * 2];      // 32 KB
    __shared__ float s_cx[KK1], s_cy[KK1], s_sx[KK1], s_sy[KK1], s_c1[KK1];

    const int tid = threadIdx.x;
    const int n   = blockIdx.x;

    // ---- stage z0 (1024 x 16B) and z1 (2048 x 16B) into LDS, async ----
    for (int i = tid; i < (BB * TT) / 4; i += 256)
        async_copy_b128(z0 + i * 4, &s_z0[i * 4]);
    for (int i = tid; i < (BB * TT * 2) / 4; i += 256)
        async_copy_b128(z1 + i * 4, &s_z1[i * 4]);

    // ---- per-neuron euclidean k-table (uniform -> scalar loads) ----
    if (tid < KK1) {
        const float rfx = rf1[2 * n + 0];
        const float rfy = rf1[2 * n + 1];
        s_cx[tid] = mean1[2 * tid + 0] - rfx;
        s_cy[tid] = mean1[2 * tid + 1] - rfy;
        s_sx[tid] = 1.0f / __expf(log_var1[2 * tid + 0]);
        s_sy[tid] = 1.0f / __expf(log_var1[2 * tid + 1]);
        s_c1[tid] = coeff1[tid];
    }

    // ---- per-neuron torus / mixture constants (uniform) ----
    const float s0  = 1.0f / __expf(log_var0[0]);
    const float c0  = coeff0[0];
    const float m0  = mean0[0];
    const float rfv = rf0[n];
    const float ax  = __sinf(m0) - __sinf(rfv);
    const float ay  = __cosf(m0) - __cosf(rfv);
    const float e0  = __expf(ew[2 * n + 0]);
    const float e1  = __expf(ew[2 * n + 1]);
    const float inv = 1.0f / (e0 + e1);
    const float w0  = e0 * inv;
    const float w1  = e1 * inv;
    const float scl = __expf(lfs[n]);

    wait_async_all();      // per-wave ASYNCcnt drain
    __syncthreads();       // all waves' DMA landed + k-table visible

    // ---- main loop: b = 0..15, thread owns t = tid (coalesced stores) ----
    const float2* __restrict__ z1v = (const float2*)s_z1;
    for (int b = 0; b < BB; ++b) {
        const int bt = b * TT + tid;

        // torus ensemble (K0 = 1)
        const float za = s_z0[bt];
        const float dx = (__sinf(za) - ax) * s0;
        const float dy = (__cosf(za) - ay) * s0;
        const float r0 = __expf(c0 * __expf(-(dx * dx + dy * dy)));

        // euclidean ensemble (K1 = 16)
        const float2 zv = z1v[bt];
        float acc = 0.0f;
#pragma unroll
        for (int k = 0; k < KK1; ++k) {
            const float ex = (zv.x - s_cx[k]) * s_sx[k];
            const float ey = (zv.y - s_cy[k]) * s_sy[k];
            acc += s_c1[k] * __expf(-(ex * ex + ey * ey));
        }
        const float r1 = __expf(acc);

        out[((size_t)b * N + n) * TT + tid] = (w0 * r0 + w1 * r1) * scl;
    }
}

extern "C" void kernel_launch(void* const* d_in, const int* in_sizes, int n_in,
                              void* d_out, int out_size, void* d_ws, size_t ws_size,
                              hipStream_t stream) {
    const float* z0       = (const float*)d_in[0];
    const float* z1       = (const float*)d_in[1];
    const float* coeff0   = (const float*)d_in[2];
    const float* mean0    = (const float*)d_in[3];
    const float* log_var0 = (const float*)d_in[4];
    const float* coeff1   = (const float*)d_in[5];
    const float* mean1    = (const float*)d_in[6];
    const float* log_var1 = (const float*)d_in[7];
    const float* rf_tr0   = (const float*)d_in[8];
    const float* rf_tr1   = (const float*)d_in[9];
    const float* rf_te0   = (const float*)d_in[10];
    const float* rf_te1   = (const float*)d_in[11];
    const float* ew_tr    = (const float*)d_in[12];
    const float* ew_te    = (const float*)d_in[13];
    const float* lfs_tr   = (const float*)d_in[14];
    const float* lfs_te   = (const float*)d_in[15];

    float* out_tr = (float*)d_out;                          // (B, NTR, T)
    float* out_te = out_tr + (size_t)BB * NTR * TT;         // (B, NTE, T)

    decoder_fused_kernel<<<NTR, 256, 0, stream>>>(
        z0, z1, coeff0, mean0, log_var0, coeff1, mean1, log_var1,
        rf_tr0, rf_tr1, ew_tr, lfs_tr, out_tr, NTR);

    decoder_fused_kernel<<<NTE, 256, 0, stream>>>(
        z0, z1, coeff0, mean0, log_var0, coeff1, mean1, log_var1,
        rf_te0, rf_te1, ew_te, lfs_te, out_te, NTE);
}